// Wave_generator_53360673685670
// MI455X (gfx1250) — compile-verified
//
#include <hip/hip_runtime.h>
#include <math.h>

// ---- problem constants ----
#define BB 8
#define NN 1024
#define HH 128
#define NHEADS 4
#define DH 32
#define NCOE 50
#define NSC 4
#define THRE 5.0f

// ---- workspace layout (float offsets) ----
#define WS_EIG      ((size_t)0)            // [B,N,H]   1048576
#define WS_X        ((size_t)1048576)      // [B,N,H]   1048576
#define WS_QKV      ((size_t)2097152)      // [B,N,384] 3145728
#define WS_O        ((size_t)5242880)      // [B,N,H]   1048576
#define WS_Y1       ((size_t)6291456)      // [B,N,H]   1048576
#define WS_EIGSUM   ((size_t)7340032)      // [B,H]     1024
#define WS_COESC    ((size_t)7341056)      // [B,50]    400
#define WS_COEWV    ((size_t)7341456)      // [B,50]    400
#define WS_COESL    ((size_t)7341856)      // [B,4]     32
#define WS_FSAT     ((size_t)7341888)      // [B,5,N]   40960 (transposed: fsaT[b][s][k])
#define WS_COLNORM  ((size_t)7382848)      // [B,5,N]   40960

typedef __attribute__((ext_vector_type(2))) float v2f;
typedef __attribute__((ext_vector_type(8))) float v8f;

// ============================================================
// 1) Sine encoding + eigw linear: eig[b,n,h]
// ============================================================
__global__ void k_sine_eigw(const float* __restrict__ ev, const float* __restrict__ W,
                            const float* __restrict__ bias, float* __restrict__ eig) {
    __shared__ float ee[16][132];
    const int rowBase = blockIdx.x * 16;
    const int t = threadIdx.x;
    const float c0 = -logf(10000.0f) / (float)HH;
    for (int idx = t; idx < 16 * 129; idx += 256) {
        int r = idx / 129, j = idx % 129;
        float lam = ev[rowBase + r];
        float val;
        if (j == 0) val = lam;
        else if (j <= 64) { float d = expf(2.0f * (float)(j - 1) * c0);  val = sinf(lam * 100.0f * d); }
        else              { float d = expf(2.0f * (float)(j - 65) * c0); val = cosf(lam * 100.0f * d); }
        ee[r][j] = val;
    }
    __syncthreads();
    int r  = t >> 4;       // 0..15
    int hg = t & 15;       // 8 outputs each
#pragma unroll
    for (int i = 0; i < 8; ++i) {
        int h = hg * 8 + i;
        float acc = bias[h];
        const float* w = W + h * 129;
        for (int k = 0; k < 129; ++k) acc += ee[r][k] * w[k];
        eig[(size_t)(rowBase + r) * HH + h] = acc;
    }
}

// ============================================================
// 2) LayerNorm: one wave per row (wave32)
// ============================================================
__global__ void k_layernorm(const float* __restrict__ in, const float* __restrict__ g,
                            const float* __restrict__ bsh, float* __restrict__ out) {
    int wave = threadIdx.x >> 5;
    int lane = threadIdx.x & 31;
    int row  = blockIdx.x * 4 + wave;
    const float* x = in + (size_t)row * HH;
    float v[4], s = 0.f, ss = 0.f;
#pragma unroll
    for (int i = 0; i < 4; ++i) { v[i] = x[lane + 32 * i]; s += v[i]; ss += v[i] * v[i]; }
    for (int off = 16; off; off >>= 1) { s += __shfl_xor(s, off, 32); ss += __shfl_xor(ss, off, 32); }
    float mu  = s / (float)HH;
    float var = ss / (float)HH - mu * mu;
    float rs  = rsqrtf(var + 1e-5f);
    float* o = out + (size_t)row * HH;
#pragma unroll
    for (int i = 0; i < 4; ++i) { int c = lane + 32 * i; o[c] = (v[i] - mu) * rs * g[c] + bsh[c]; }
}

// ============================================================
// 3) Generic linear: out[M,Nout] = act(A[M,128] @ W[Nout,128]^T + b) (+res)
// ============================================================
template <int ACT, bool RES>
__global__ void k_linear(const float* __restrict__ A, const float* __restrict__ W,
                         const float* __restrict__ bias, const float* __restrict__ res,
                         float* __restrict__ out, int Nout) {
    __shared__ float As[32][132];
    __shared__ float Ws[64][132];
    int t = threadIdx.x;
    int rowBase = blockIdx.x * 32;
    int colBase = blockIdx.y * 64;
    for (int idx = t; idx < 32 * 128; idx += 256) {
        int r = idx >> 7, k = idx & 127;
        As[r][k] = A[(size_t)(rowBase + r) * HH + k];
    }
    for (int idx = t; idx < 64 * 128; idx += 256) {
        int c = idx >> 7, k = idx & 127;
        Ws[c][k] = W[(size_t)(colBase + c) * HH + k];
    }
    __syncthreads();
    int r = t >> 3, cg = t & 7;
    float acc[8];
#pragma unroll
    for (int i = 0; i < 8; ++i) acc[i] = 0.f;
    for (int k = 0; k < 128; ++k) {
        float a = As[r][k];
#pragma unroll
        for (int i = 0; i < 8; ++i) acc[i] += a * Ws[cg * 8 + i][k];
    }
    int grow = rowBase + r;
#pragma unroll
    for (int i = 0; i < 8; ++i) {
        int c = colBase + cg * 8 + i;
        float v = acc[i] + bias[c];
        if (ACT == 1) v = 0.5f * v * (1.0f + erff(v * 0.70710678118654752f));
        if (RES) v += res[(size_t)grow * Nout + c];
        out[(size_t)grow * Nout + c] = v;
    }
}

// ============================================================
// 4) Flash attention: 8 waves per block share (b,h); wave = one query row
// ============================================================
__global__ void k_attention(const float* __restrict__ qkv, const int* __restrict__ length,
                            float* __restrict__ o) {
    __shared__ float Ks[32][33];
    __shared__ float Vs[32][33];
    __shared__ float Qs[8][33];
    int t = threadIdx.x;
    int bh = blockIdx.x >> 7;                 // (N/8)=128 blocks per (b,h)
    int qBase = (blockIdx.x & 127) * 8;
    int b = bh >> 2, h = bh & 3;
    int len = length[b];
    const float scale = 0.1767766952966369f;  // 1/sqrt(32)
    {
        int r = t >> 5, d = t & 31;
        Qs[r][d] = qkv[((size_t)(b * NN + qBase + r)) * (3 * HH) + h * DH + d] * scale;
    }
    int wave = t >> 5, lane = t & 31;
    float m = -INFINITY, l = 0.f, acc = 0.f;
    for (int j0 = 0; j0 < NN; j0 += 32) {
        __syncthreads();
        for (int idx = t; idx < 32 * 32; idx += 256) {
            int jj = idx >> 5, d = idx & 31;
            size_t base = ((size_t)(b * NN + j0 + jj)) * (3 * HH);
            Ks[jj][d] = qkv[base + HH + h * DH + d];
            Vs[jj][d] = qkv[base + 2 * HH + h * DH + d];
        }
        __syncthreads();
        float s = 0.f;
#pragma unroll 8
        for (int d = 0; d < 32; ++d) s += Qs[wave][d] * Ks[lane][d];
        if (j0 + lane >= len) s = -1e9f;
        float cmax = s;
        for (int off = 16; off; off >>= 1) cmax = fmaxf(cmax, __shfl_xor(cmax, off, 32));
        float mn = fmaxf(m, cmax);
        float corr = expf(m - mn);
        float p = expf(s - mn);
        float ps = p;
        for (int off = 16; off; off >>= 1) ps += __shfl_xor(ps, off, 32);
        l = l * corr + ps;
        acc *= corr;
        for (int jj = 0; jj < 32; ++jj) {
            float pb = __shfl(p, jj, 32);
            acc += pb * Vs[jj][lane];
        }
        m = mn;
    }
    o[(size_t)(b * NN + qBase + wave) * HH + h * DH + lane] = acc / l;
}

// ============================================================
// 5) Masked row-sum over nodes: eigsum[b,h]
// ============================================================
__global__ void k_rowsum(const float* __restrict__ eig, const int* __restrict__ length,
                         float* __restrict__ eigsum) {
    int b = blockIdx.x, h = threadIdx.x;
    int len = length[b]; if (len > NN) len = NN;
    float s = 0.f;
    for (int n = 0; n < len; ++n) s += eig[(size_t)(b * NN + n) * HH + h];
    eigsum[b * HH + h] = s;
}

// ============================================================
// 6) Decoder coefficients: sigmoid-pool + normalize
// ============================================================
__global__ void k_coeffs(const float* __restrict__ eigsum, const int* __restrict__ length,
                         const float* __restrict__ dscW, const float* __restrict__ dscB,
                         const float* __restrict__ dwavW, const float* __restrict__ dwavB,
                         const float* __restrict__ dsclW, const float* __restrict__ dsclB,
                         float* __restrict__ coeSc, float* __restrict__ coeWv,
                         float* __restrict__ coeSl) {
    __shared__ float sv[NCOE];
    __shared__ float ssum;
    int b = blockIdx.x, j = threadIdx.x;
    float leni = (float)length[b];
    float lenf = leni + 1e-8f;
    const float* es = eigsum + b * HH;
    if (j < NCOE) {
        float a = 0.f;
        for (int k = 0; k < HH; ++k) a += es[k] * dscW[j * HH + k];
        float pooled = (a + leni * dscB[j]) / lenf;
        sv[j] = 1.f / (1.f + expf(-pooled));
    }
    __syncthreads();
    if (j == 0) { float s = 0.f; for (int c = 0; c < NCOE; ++c) s += sv[c]; ssum = s + 1e-8f; }
    __syncthreads();
    if (j < NCOE) coeSc[b * NCOE + j] = sv[j] / ssum;
    __syncthreads();
    if (j < NCOE) {
        float a = 0.f;
        for (int k = 0; k < HH; ++k) a += es[k] * dwavW[j * HH + k];
        float pooled = (a + leni * dwavB[j]) / lenf;
        sv[j] = 1.f / (1.f + expf(-pooled));
    }
    __syncthreads();
    if (j == 0) { float s = 0.f; for (int c = 0; c < NCOE; ++c) s += sv[c]; ssum = s + 1e-8f; }
    __syncthreads();
    if (j < NCOE) coeWv[b * NCOE + j] = sv[j] / ssum;
    if (j < NSC) {
        float a = 0.f;
        for (int k = 0; k < HH; ++k) a += es[k] * dsclW[j * HH + k];
        float pooled = (a + leni * dsclB[j]) / lenf;
        coeSl[b * NSC + j] = THRE / (1.f + expf(-pooled));
    }
}

// ============================================================
// 7) Chebyshev recurrences + tight-frame normalize -> fsaT[B,5,N]
//    (transposed layout so the GEMM can fetch scales as float4)
// ============================================================
__global__ void k_fsa(const float* __restrict__ ev, const float* __restrict__ coeSc,
                      const float* __restrict__ coeWv, const float* __restrict__ coeSl,
                      float* __restrict__ fsaT) {
    __shared__ float sc[NCOE], wv[NCOE], sl[NSC];
    int t = threadIdx.x;
    int gid = blockIdx.x * 256 + t;
    int b = gid >> 10;
    int n = gid & (NN - 1);
    if (t < NCOE) { sc[t] = coeSc[b * NCOE + t]; wv[t] = coeWv[b * NCOE + t]; }
    if (t < NSC) sl[t] = coeSl[b * NSC + t];
    __syncthreads();
    float lam = ev[gid];
    float vals[5];
    {   // scaling: odd Chebyshev polys of (lam-1)
        float y = lam - 1.0f;
        float te = 1.0f, to = y;
        float cs = sc[0] * 0.5f * (1.0f - to);
        for (int c = 1; c < NCOE; ++c) {
            te = 2.0f * y * to - te;
            to = 2.0f * y * te - to;
            cs += sc[c] * 0.5f * (1.0f - to);
        }
        vals[0] = cs;
    }
    for (int sI = 0; sI < NSC; ++sI) {   // wavelet: even Chebyshev polys of (fsw-1)
        float f = lam * sl[sI];
        if (f > 2.0f) f = 0.0f;
        float y = f - 1.0f;
        float te = 1.0f, to = y;
        float cw = 0.0f;                 // c=0 term is 0.5*(1-T0)=0
        for (int c = 1; c < NCOE; ++c) {
            te = 2.0f * y * to - te;
            to = 2.0f * y * te - to;
            cw += wv[c] * 0.5f * (1.0f - te);
        }
        vals[1 + sI] = cw;
    }
    float nrm = 0.f;
#pragma unroll
    for (int i = 0; i < 5; ++i) nrm += vals[i] * vals[i];
    nrm = sqrtf(nrm) + 1e-8f;
#pragma unroll
    for (int i = 0; i < 5; ++i)
        fsaT[((size_t)(b * 5 + i)) * NN + n] = vals[i] / nrm;
}

// ============================================================
// 8) Filters GEMM with V_WMMA_F32_16X16X4_F32, double-buffered LDS.
//    C[n,m] = sum_k evc[n,k]*fsa[k,s]*evc[m,k]; 128x128 tile / 256 threads.
//    Chunk c+1 is prefetched to registers while chunk c runs on the
//    matrix pipe; one barrier per k-chunk.
// ============================================================
__global__ void __launch_bounds__(256)
k_gemm_filters(const float* __restrict__ evc, const float* __restrict__ fsaT,
               float* __restrict__ bases, float* __restrict__ colnorm) {
    __shared__ float As[2][128][17];
    __shared__ float Bs[2][128][17];
    int t = threadIdx.x;
    int p = blockIdx.y;             // 0..39
    int b = p / 5, s = p % 5;
    int tx = blockIdx.x & 7;        // col macro-tile
    int ty = blockIdx.x >> 3;       // row macro-tile
    const float* E = evc + (size_t)b * NN * NN;
    const float* F = fsaT + ((size_t)(b * 5 + s)) * NN;
    int wave = t >> 5, lane = t & 31;
    int wr = wave >> 2, wc = wave & 3;
    int lm = lane & 15, hi = lane >> 4;

    // per-thread tile-load coordinates (2 quads each for A and B)
    int lr[2], lq[2];
#pragma unroll
    for (int i = 0; i < 2; ++i) {
        int idx = t + i * 256;      // 0..511
        lr[i] = idx >> 2;           // 0..127
        lq[i] = (idx & 3) * 4;      // 0,4,8,12
    }

    v8f acc[4][2];
#pragma unroll
    for (int i = 0; i < 4; ++i)
#pragma unroll
        for (int j = 0; j < 2; ++j) acc[i][j] = (v8f)(0.0f);

    float4 ar[2], br[2], fr[2];
    auto loadChunk = [&](int k0) {
#pragma unroll
        for (int i = 0; i < 2; ++i) {
            ar[i] = *(const float4*)(E + (size_t)(ty * 128 + lr[i]) * NN + k0 + lq[i]);
            br[i] = *(const float4*)(E + (size_t)(tx * 128 + lr[i]) * NN + k0 + lq[i]);
            fr[i] = *(const float4*)(F + k0 + lq[i]);
        }
    };
    auto storeChunk = [&](int buf) {
#pragma unroll
        for (int i = 0; i < 2; ++i) {
            int r = lr[i], q = lq[i];
            As[buf][r][q + 0] = ar[i].x * fr[i].x;
            As[buf][r][q + 1] = ar[i].y * fr[i].y;
            As[buf][r][q + 2] = ar[i].z * fr[i].z;
            As[buf][r][q + 3] = ar[i].w * fr[i].w;
            Bs[buf][r][q + 0] = br[i].x;
            Bs[buf][r][q + 1] = br[i].y;
            Bs[buf][r][q + 2] = br[i].z;
            Bs[buf][r][q + 3] = br[i].w;
        }
    };

    loadChunk(0);
    storeChunk(0);
    __syncthreads();

    const int NCHUNK = NN / 16;     // 64
    for (int c = 0; c < NCHUNK; ++c) {
        int cur = c & 1;
        if (c + 1 < NCHUNK) loadChunk((c + 1) * 16);   // prefetch next chunk (overlaps WMMA)
#pragma unroll
        for (int kk = 0; kk < 16; kk += 4) {
            int kb = kk + hi * 2;
            v2f bf[2], af[4];
#pragma unroll
            for (int j = 0; j < 2; ++j) {
                int nIdx = wc * 32 + j * 16 + lm;
                bf[j].x = Bs[cur][nIdx][kb];
                bf[j].y = Bs[cur][nIdx][kb + 1];
            }
#pragma unroll
            for (int i = 0; i < 4; ++i) {
                int mIdx = wr * 64 + i * 16 + lm;
                af[i].x = As[cur][mIdx][kb];
                af[i].y = As[cur][mIdx][kb + 1];
            }
#pragma unroll
            for (int i = 0; i < 4; ++i)
#pragma unroll
                for (int j = 0; j < 2; ++j)
                    acc[i][j] = __builtin_amdgcn_wmma_f32_16x16x4_f32(
                        false, af[i], false, bf[j], (short)0, acc[i][j], false, false);
        }
        if (c + 1 < NCHUNK) storeChunk(cur ^ 1);
        __syncthreads();
    }

    size_t outBase = (size_t)s * BB * NN * NN + (size_t)b * NN * NN;
#pragma unroll
    for (int i = 0; i < 4; ++i)
#pragma unroll
        for (int j = 0; j < 2; ++j) {
            int row0 = ty * 128 + wr * 64 + i * 16;
            int col0 = tx * 128 + wc * 32 + j * 16;
            int n = lm;
            float ssq = 0.f;
#pragma unroll
            for (int v = 0; v < 8; ++v) {
                float val = acc[i][j][v];
                bases[outBase + (size_t)(row0 + v + hi * 8) * NN + col0 + n] = val;
                ssq += val * val;
            }
            atomicAdd(&colnorm[(b * 5 + s) * NN + col0 + n], ssq);
        }
}

// ============================================================
// 9) Column normalization (F.normalize over n), float4 vectorized
// ============================================================
__global__ void k_normalize(float* __restrict__ bases, const float* __restrict__ colnorm) {
    size_t idx4 = (size_t)blockIdx.x * 256 + threadIdx.x;
    size_t e = idx4 * 4;
    int m = (int)(e & (NN - 1));
    size_t sIdx = e / ((size_t)BB * NN * NN);
    size_t rem = e % ((size_t)BB * NN * NN);
    int b = (int)(rem / ((size_t)NN * NN));
    const float4 nv = *(const float4*)(colnorm + ((size_t)b * 5 + sIdx) * NN + m);
    float4 v = *(float4*)(bases + e);
    v.x /= fmaxf(sqrtf(nv.x), 1e-12f);
    v.y /= fmaxf(sqrtf(nv.y), 1e-12f);
    v.z /= fmaxf(sqrtf(nv.z), 1e-12f);
    v.w /= fmaxf(sqrtf(nv.w), 1e-12f);
    *(float4*)(bases + e) = v;
}

// ============================================================
// launch
// ============================================================
extern "C" void kernel_launch(void* const* d_in, const int* in_sizes, int n_in,
                              void* d_out, int out_size, void* d_ws, size_t ws_size,
                              hipStream_t stream) {
    const float* eigenvalue  = (const float*)d_in[0];
    const float* eigenvector = (const float*)d_in[1];
    const float* eigw_W = (const float*)d_in[2];
    const float* eigw_b = (const float*)d_in[3];
    const float* ln1_g  = (const float*)d_in[4];
    const float* ln1_b  = (const float*)d_in[5];
    const float* qkv_W  = (const float*)d_in[6];
    const float* qkv_b  = (const float*)d_in[7];
    const float* out_W  = (const float*)d_in[8];
    const float* out_b  = (const float*)d_in[9];
    const float* ln2_g  = (const float*)d_in[10];
    const float* ln2_b  = (const float*)d_in[11];
    const float* ffn1_W = (const float*)d_in[12];
    const float* ffn1_b = (const float*)d_in[13];
    const float* ffn2_W = (const float*)d_in[14];
    const float* ffn2_b = (const float*)d_in[15];
    const float* dsc_W  = (const float*)d_in[16];
    const float* dsc_b  = (const float*)d_in[17];
    const float* dwav_W = (const float*)d_in[18];
    const float* dwav_b = (const float*)d_in[19];
    const float* dscl_W = (const float*)d_in[20];
    const float* dscl_b = (const float*)d_in[21];
    const int*   length = (const int*)d_in[22];

    float* ws = (float*)d_ws;
    float* eig    = ws + WS_EIG;
    float* x      = ws + WS_X;
    float* qkv    = ws + WS_QKV;
    float* attn_o = ws + WS_O;
    float* y1     = ws + WS_Y1;
    float* eigsum = ws + WS_EIGSUM;
    float* coeSc  = ws + WS_COESC;
    float* coeWv  = ws + WS_COEWV;
    float* coeSl  = ws + WS_COESL;
    float* fsaT   = ws + WS_FSAT;
    float* coln   = ws + WS_COLNORM;
    float* bases  = (float*)d_out;

    hipMemsetAsync(coln, 0, (size_t)BB * 5 * NN * sizeof(float), stream);

    k_sine_eigw<<<512, 256, 0, stream>>>(eigenvalue, eigw_W, eigw_b, eig);
    k_layernorm<<<2048, 128, 0, stream>>>(eig, ln1_g, ln1_b, x);
    k_linear<0, false><<<dim3(256, 6), 256, 0, stream>>>(x, qkv_W, qkv_b, nullptr, qkv, 384);
    k_attention<<<4096, 256, 0, stream>>>(qkv, length, attn_o);
    k_linear<0, true><<<dim3(256, 2), 256, 0, stream>>>(attn_o, out_W, out_b, eig, eig, 128);
    k_layernorm<<<2048, 128, 0, stream>>>(eig, ln2_g, ln2_b, x);
    k_linear<1, false><<<dim3(256, 2), 256, 0, stream>>>(x, ffn1_W, ffn1_b, nullptr, y1, 128);
    k_linear<0, true><<<dim3(256, 2), 256, 0, stream>>>(y1, ffn2_W, ffn2_b, eig, eig, 128);
    k_rowsum<<<BB, HH, 0, stream>>>(eig, length, eigsum);
    k_coeffs<<<BB, 64, 0, stream>>>(eigsum, length, dsc_W, dsc_b, dwav_W, dwav_b,
                                    dscl_W, dscl_b, coeSc, coeWv, coeSl);
    k_fsa<<<32, 256, 0, stream>>>(eigenvalue, coeSc, coeWv, coeSl, fsaT);
    k_gemm_filters<<<dim3(64, 40), 256, 0, stream>>>(eigenvector, fsaT, bases, coln);
    k_normalize<<<40960, 256, 0, stream>>>(bases, coln);
}